// AdultConnectomeNetwork_51625506898128
// MI455X (gfx1250) — compile-verified
//
#include <hip/hip_runtime.h>
#include <stdint.h>

// ---------------------------------------------------------------------------
// AdultConnectomeNetwork: 3 layers of xt = A_sp @ (W_sp @ xt) + bias
// COO SpMM, N=50000, NNZ=1.6M, B=32 (== wave32 lanes), f32.
// Memory-bound: dense [N,32] (6.4MB) lives in L2; edges streamed via async
// global->LDS (ASYNCcnt); scatter via global_atomic_add_f32 (no return).
// ---------------------------------------------------------------------------

#define TILE_EDGES 1024
#define SPMM_BLOCK 256
#define BATCH      32

typedef int v4i __attribute__((ext_vector_type(4)));
typedef __attribute__((address_space(1))) v4i gas_v4i;  // global (AS1)
typedef __attribute__((address_space(3))) v4i las_v4i;  // LDS    (AS3)

__device__ __forceinline__ void async_copy_b128(const void* gsrc, void* ldst) {
#if __has_builtin(__builtin_amdgcn_global_load_async_to_lds_b128)
  __builtin_amdgcn_global_load_async_to_lds_b128(
      (gas_v4i*)gsrc, (las_v4i*)ldst, /*offset=*/0, /*cpol=*/0);
#else
  unsigned lofs = (unsigned)(size_t)(las_v4i*)ldst;
  asm volatile("global_load_async_to_lds_b128 %0, %1, off"
               :: "v"(lofs), "v"(gsrc) : "memory");
#endif
}

__device__ __forceinline__ void wait_async_zero() {
#if __has_builtin(__builtin_amdgcn_s_wait_asynccnt)
  __builtin_amdgcn_s_wait_asynccnt(0);
#else
  asm volatile("s_wait_asynccnt 0x0" ::: "memory");
#endif
}

__device__ __forceinline__ void atomic_add_f32(float* p, float v) {
  // HIP fast-path f32 atomic -> global_atomic_add_f32 (no return, STOREcnt)
  unsafeAtomicAdd(p, v);
}

// Full tiles only (branch-free): y[row*32+lane] += vals * x[col*32+lane],
// 1024 edges per block staged through LDS via async copies.
__global__ void __launch_bounds__(SPMM_BLOCK)
spmm_edges_full(const float* __restrict__ vals,
                const int*   __restrict__ row,
                const int*   __restrict__ col,
                const float* __restrict__ xin,   // [N,32]
                float*       __restrict__ yout)  // [N,32], pre-initialized
{
  __shared__ float s_vals[TILE_EDGES];
  __shared__ int   s_row[TILE_EDGES];
  __shared__ int   s_col[TILE_EDGES];

  const int tid  = threadIdx.x;
  const long long base = (long long)blockIdx.x * TILE_EDGES;

  // Stage 1024 edges: each thread async-copies 16B (4 edges) per array.
  const int c4 = tid * 4;
  async_copy_b128(vals + base + c4, &s_vals[c4]);
  async_copy_b128(row  + base + c4, &s_row[c4]);
  async_copy_b128(col  + base + c4, &s_col[c4]);
  wait_async_zero();       // my async copies landed in LDS
  __syncthreads();         // everyone's copies landed

  // Each wave processes 128 edges; all 32 lanes share one edge (broadcast
  // LDS reads, wave-uniform address), lane = batch column -> coalesced
  // 128B gather + coalesced 128B atomic scatter. Unroll 8 so the compiler
  // batches gathers (ds_load_b128 x3 per 4 edges) ahead of the atomics.
  // All dense indices fit in u32 -> SADDR + 32-bit voffset addressing.
  const unsigned lane = tid & 31;
  const unsigned wave = tid >> 5;               // 0..7
  const unsigned ebeg = wave * (TILE_EDGES / 8);
#pragma unroll 8
  for (unsigned i = 0; i < TILE_EDGES / 8; ++i) {
    const unsigned el = ebeg + i;
    const float    v  = s_vals[el];
    const unsigned r  = (unsigned)s_row[el];
    const unsigned c  = (unsigned)s_col[el];
    const float    xv = xin[c * BATCH + lane];
    atomic_add_f32(&yout[r * BATCH + lane], v * xv);
  }
}

// Remainder edges (< TILE_EDGES): one block, direct bounds-checked loads.
__global__ void __launch_bounds__(SPMM_BLOCK)
spmm_edges_tail(const float* __restrict__ vals,
                const int*   __restrict__ row,
                const int*   __restrict__ col,
                const float* __restrict__ xin,
                float*       __restrict__ yout,
                int start, int nnz)
{
  const unsigned lane = threadIdx.x & 31;
  const unsigned wave = threadIdx.x >> 5;
  for (int e = start + (int)wave; e < nnz; e += (SPMM_BLOCK / 32)) {
    const float    v = vals[e];
    const unsigned r = (unsigned)row[e];
    const unsigned c = (unsigned)col[e];
    atomic_add_f32(&yout[r * BATCH + lane], v * xin[c * BATCH + lane]);
  }
}

__global__ void init_zero4(float* __restrict__ p, int n4) {
  int i = blockIdx.x * blockDim.x + threadIdx.x;
  if (i < n4) reinterpret_cast<float4*>(p)[i] = make_float4(0.f, 0.f, 0.f, 0.f);
}

// p[n*32 + b] = bias[n]  (accumulator init; SpMM atomics add on top)
__global__ void init_bias(float* __restrict__ p, const float* __restrict__ bias, int n) {
  unsigned gid = blockIdx.x * blockDim.x + threadIdx.x;
  if (gid < (unsigned)(n * BATCH)) p[gid] = bias[gid >> 5];
}

// xt[n*32 + b] = x[b*N + n]
__global__ void transpose_bn_to_nb(const float* __restrict__ x,
                                   float* __restrict__ xt, int n) {
  unsigned gid = blockIdx.x * blockDim.x + threadIdx.x;
  if (gid < (unsigned)(n * BATCH)) {
    unsigned nn = gid >> 5, b = gid & 31;
    xt[gid] = x[b * (unsigned)n + nn];
  }
}

// out[b*N + n] = xt[n*32 + b]
__global__ void transpose_nb_to_bn(const float* __restrict__ xt,
                                   float* __restrict__ out, int n) {
  unsigned gid = blockIdx.x * blockDim.x + threadIdx.x;
  if (gid < (unsigned)(n * BATCH)) {
    unsigned b = gid / (unsigned)n, nn = gid % (unsigned)n;
    out[gid] = xt[nn * BATCH + b];
  }
}

static inline void launch_spmm(const float* vals, const int* row, const int* col,
                               const float* xin, float* yout,
                               int nnz, int full_tiles, hipStream_t stream) {
  if (full_tiles > 0) {
    spmm_edges_full<<<full_tiles, SPMM_BLOCK, 0, stream>>>(vals, row, col, xin, yout);
  }
  const int rem_start = full_tiles * TILE_EDGES;
  if (rem_start < nnz) {
    spmm_edges_tail<<<1, SPMM_BLOCK, 0, stream>>>(vals, row, col, xin, yout,
                                                  rem_start, nnz);
  }
}

extern "C" void kernel_launch(void* const* d_in, const int* in_sizes, int n_in,
                              void* d_out, int out_size, void* d_ws, size_t ws_size,
                              hipStream_t stream) {
  const float* x        = (const float*)d_in[0];  // [B, N]
  const float* adj_vals = (const float*)d_in[1];  // [NNZ]
  const float* w_vals   = (const float*)d_in[2];  // [NNZ]
  const float* bias     = (const float*)d_in[3];  // [N]
  const int*   row      = (const int*)d_in[4];    // [NNZ]
  const int*   col      = (const int*)d_in[5];    // [NNZ]

  const int nnz = in_sizes[1];
  const int n   = in_sizes[3];
  const int nb  = n * BATCH;

  float* xt0 = (float*)d_ws;                  // [N,32] accumulator A-side
  float* xt1 = xt0 + (size_t)nb;              // [N,32] accumulator W-side

  const int tgrid      = (nb + 255) / 256;
  const int zgrid      = (nb / 4 + 255) / 256;
  const int full_tiles = nnz / TILE_EDGES;

  transpose_bn_to_nb<<<tgrid, 256, 0, stream>>>(x, xt0, n);

  for (int l = 0; l < 3; ++l) {
    // xt1 = W_sp @ xt0
    init_zero4<<<zgrid, 256, 0, stream>>>(xt1, nb / 4);
    launch_spmm(w_vals, row, col, xt0, xt1, nnz, full_tiles, stream);
    // xt0 = A_sp @ xt1 + bias  (bias folded into accumulator init)
    init_bias<<<tgrid, 256, 0, stream>>>(xt0, bias, n);
    launch_spmm(adj_vals, row, col, xt1, xt0, nnz, full_tiles, stream);
  }

  transpose_nb_to_bn<<<tgrid, 256, 0, stream>>>(xt0, (float*)d_out, n);
}